// GAT_46505905881799
// MI455X (gfx1250) — compile-verified
//
#include <hip/hip_runtime.h>
#include <hip/hip_bf16.h>

typedef __attribute__((ext_vector_type(2))) float v2f;
typedef __attribute__((ext_vector_type(8))) float v8f;

#define N_NODES 50000
#define N_EDGES 800000
#define E_TOT   (N_EDGES + N_NODES)   // with self loops
#define IN_DIM  128
#define HID     32
#define HEADS   4
#define D1      (HEADS * HID)         // 128
#define OUT_DIM 40
#define NEG_SLOPE 0.2f

// ---- monotonic float<->uint encoding for atomic max ----
__device__ __forceinline__ unsigned fenc(float f) {
    unsigned u = __float_as_uint(f);
    return (u & 0x80000000u) ? ~u : (u | 0x80000000u);
}
__device__ __forceinline__ float fdec(unsigned u) {
    u = (u & 0x80000000u) ? (u & 0x7fffffffu) : ~u;
    return __uint_as_float(u);
}
#define ENC_NEG_INF 0x007FFFFFu   // fenc(-inf)

// =====================  init kernels  =====================
__global__ void init1_kernel(float* out1, const float* b1, unsigned* nmax1, float* denom1) {
    int i = blockIdx.x * blockDim.x + threadIdx.x;
    if (i < N_NODES * D1) out1[i] = b1[i % D1];
    if (i < N_NODES * HEADS) { nmax1[i] = ENC_NEG_INF; denom1[i] = 0.0f; }
}
__global__ void init2_kernel(float* dout, const float* b2, unsigned* nmax2, float* denom2) {
    int i = blockIdx.x * blockDim.x + threadIdx.x;
    if (i < N_NODES * OUT_DIM) dout[i] = b2[i % OUT_DIM];
    if (i < N_NODES) { nmax2[i] = ENC_NEG_INF; denom2[i] = 0.0f; }
}

// =====================  GEMM 1: H1 = X @ W1  (fp32 WMMA) =====================
// 8 waves / block, each wave owns one 16x16 tile; block covers 16 rows x 128 cols.
__global__ __launch_bounds__(256) void gemm1_wmma(const float* __restrict__ X,
                                                  const float* __restrict__ W,
                                                  float* __restrict__ H) {
    const int wave = threadIdx.x >> 5;
    const int lane = threadIdx.x & 31;
    const int half = lane >> 4;        // 0: K=0,1  1: K=2,3 (A/B layout, ISA 7.12.2)
    const int idx  = lane & 15;
    const int row0 = blockIdx.x * 16;
    const int col0 = wave * 16;
    const float* arow = X + (size_t)(row0 + idx) * IN_DIM;
    v8f acc = {};
#pragma unroll
    for (int k = 0; k < IN_DIM; k += 4) {
        const int ka = k + 2 * half;
        v2f a, b;
        a.x = arow[ka];
        a.y = arow[ka + 1];
        b.x = W[(size_t)ka * D1 + col0 + idx];
        b.y = W[(size_t)(ka + 1) * D1 + col0 + idx];
        acc = __builtin_amdgcn_wmma_f32_16x16x4_f32(false, a, false, b,
                                                    (short)0, acc, false, false);
    }
    // C/D layout: vgpr r -> row (r + 8*half), col idx
    float* out = H + (size_t)row0 * D1 + col0;
#pragma unroll
    for (int r = 0; r < 8; ++r)
        out[(size_t)(r + 8 * half) * D1 + idx] = acc[r];
}

// =====================  GEMM 2: H2 = relu(OUT1) @ W2  (fp32 WMMA) =====================
// N = 40 padded to 48 -> 3 waves / block (one 16-col tile each).
__global__ __launch_bounds__(96) void gemm2_wmma(const float* __restrict__ A,
                                                 const float* __restrict__ W,
                                                 float* __restrict__ H) {
    const int wave = threadIdx.x >> 5;
    const int lane = threadIdx.x & 31;
    const int half = lane >> 4;
    const int idx  = lane & 15;
    const int row0 = blockIdx.x * 16;
    const int col  = wave * 16 + idx;
    const float* arow = A + (size_t)(row0 + idx) * D1;
    v8f acc = {};
#pragma unroll
    for (int k = 0; k < D1; k += 4) {
        const int ka = k + 2 * half;
        v2f a, b;
        a.x = fmaxf(arow[ka], 0.0f);       // fused ReLU on A operand
        a.y = fmaxf(arow[ka + 1], 0.0f);
        b.x = (col < OUT_DIM) ? W[(size_t)ka * OUT_DIM + col] : 0.0f;
        b.y = (col < OUT_DIM) ? W[(size_t)(ka + 1) * OUT_DIM + col] : 0.0f;
        acc = __builtin_amdgcn_wmma_f32_16x16x4_f32(false, a, false, b,
                                                    (short)0, acc, false, false);
    }
    if (col < OUT_DIM) {
#pragma unroll
        for (int r = 0; r < 8; ++r)
            H[(size_t)(row0 + r + 8 * half) * OUT_DIM + col] = acc[r];
    }
}

// =====================  attention coefficients =====================
__global__ void alpha1_kernel(const float* __restrict__ H, const float* __restrict__ as_,
                              const float* __restrict__ ad_, float* asrc, float* adst) {
    int i = blockIdx.x * blockDim.x + threadIdx.x;  // node*HEADS + head
    if (i >= N_NODES * HEADS) return;
    int n = i >> 2, h = i & 3;
    const float* hp = H + (size_t)n * D1 + h * HID;
    const float* sp = as_ + h * HID;
    const float* dp = ad_ + h * HID;
    float s = 0.f, d = 0.f;
#pragma unroll
    for (int c = 0; c < HID; ++c) { s += hp[c] * sp[c]; d += hp[c] * dp[c]; }
    asrc[i] = s; adst[i] = d;
}
__global__ void alpha2_kernel(const float* __restrict__ H, const float* __restrict__ as_,
                              const float* __restrict__ ad_, float* asrc, float* adst) {
    int n = blockIdx.x * blockDim.x + threadIdx.x;
    if (n >= N_NODES) return;
    const float* hp = H + (size_t)n * OUT_DIM;
    float s = 0.f, d = 0.f;
#pragma unroll
    for (int c = 0; c < OUT_DIM; ++c) { s += hp[c] * as_[c]; d += hp[c] * ad_[c]; }
    asrc[n] = s; adst[n] = d;
}

// =====================  layer 1 edge pipeline =====================
__device__ __forceinline__ void edge_sd(const int* ei, int e, int& s, int& d) {
    if (e < N_EDGES) { s = ei[e]; d = ei[N_EDGES + e]; }
    else             { s = d = e - N_EDGES; }
}

__global__ void edge_logits1(const int* __restrict__ ei, const float* __restrict__ asrc,
                             const float* __restrict__ adst, float* elog, unsigned* nmax) {
    int e = blockIdx.x * blockDim.x + threadIdx.x;
    if (e >= E_TOT) return;
    int s, d; edge_sd(ei, e, s, d);
#pragma unroll
    for (int h = 0; h < HEADS; ++h) {
        float l = asrc[s * HEADS + h] + adst[d * HEADS + h];
        l = (l > 0.f) ? l : NEG_SLOPE * l;
        elog[(size_t)e * HEADS + h] = l;
        atomicMax(&nmax[d * HEADS + h], fenc(l));
    }
}
__global__ void edge_exp1(const int* __restrict__ ei, const unsigned* __restrict__ nmax,
                          float* elog, float* denom) {
    int e = blockIdx.x * blockDim.x + threadIdx.x;
    if (e >= E_TOT) return;
    int s, d; edge_sd(ei, e, s, d); (void)s;
#pragma unroll
    for (int h = 0; h < HEADS; ++h) {
        float ex = __expf(elog[(size_t)e * HEADS + h] - fdec(nmax[d * HEADS + h]));
        elog[(size_t)e * HEADS + h] = ex;
        atomicAdd(&denom[d * HEADS + h], ex);
    }
}
// one thread per (edge, 4-channel chunk): 32 threads/edge cover 128 channels
__global__ void edge_msg1(const int* __restrict__ ei, const float* __restrict__ H,
                          const float* __restrict__ elog, const float* __restrict__ denom,
                          float* out1) {
    long long gid = (long long)blockIdx.x * blockDim.x + threadIdx.x;
    if (gid >= (long long)E_TOT * 32) return;
    int e = (int)(gid >> 5);
    int j = (int)(gid & 31);
    int h = j >> 3;                    // head = (4j)/32
    int c0 = j << 2;                   // channel base
    int s, d; edge_sd(ei, e, s, d);
    float alpha = elog[(size_t)e * HEADS + h] / denom[d * HEADS + h];
    const float4 hv = *(const float4*)(H + (size_t)s * D1 + c0);
    float* ob = out1 + (size_t)d * D1 + c0;
    atomicAdd(ob + 0, hv.x * alpha);
    atomicAdd(ob + 1, hv.y * alpha);
    atomicAdd(ob + 2, hv.z * alpha);
    atomicAdd(ob + 3, hv.w * alpha);
}

// =====================  layer 2 edge pipeline (H=1, C=40) =====================
__global__ void edge_logits2(const int* __restrict__ ei, const float* __restrict__ asrc,
                             const float* __restrict__ adst, float* elog, unsigned* nmax) {
    int e = blockIdx.x * blockDim.x + threadIdx.x;
    if (e >= E_TOT) return;
    int s, d; edge_sd(ei, e, s, d);
    float l = asrc[s] + adst[d];
    l = (l > 0.f) ? l : NEG_SLOPE * l;
    elog[e] = l;
    atomicMax(&nmax[d], fenc(l));
}
__global__ void edge_exp2(const int* __restrict__ ei, const unsigned* __restrict__ nmax,
                          float* elog, float* denom) {
    int e = blockIdx.x * blockDim.x + threadIdx.x;
    if (e >= E_TOT) return;
    int s, d; edge_sd(ei, e, s, d); (void)s;
    float ex = __expf(elog[e] - fdec(nmax[d]));
    elog[e] = ex;
    atomicAdd(&denom[d], ex);
}
__global__ void edge_msg2(const int* __restrict__ ei, const float* __restrict__ H,
                          const float* __restrict__ elog, const float* __restrict__ denom,
                          float* dout) {
    long long gid = (long long)blockIdx.x * blockDim.x + threadIdx.x;
    if (gid >= (long long)E_TOT * OUT_DIM) return;
    int e = (int)(gid / OUT_DIM);
    int c = (int)(gid % OUT_DIM);
    int s, d; edge_sd(ei, e, s, d);
    float alpha = elog[e] / denom[d];
    atomicAdd(&dout[(size_t)d * OUT_DIM + c], H[(size_t)s * OUT_DIM + c] * alpha);
}

// =====================  launcher  =====================
extern "C" void kernel_launch(void* const* d_in, const int* in_sizes, int n_in,
                              void* d_out, int out_size, void* d_ws, size_t ws_size,
                              hipStream_t stream) {
    (void)in_sizes; (void)n_in; (void)out_size; (void)ws_size;
    const float* x        = (const float*)d_in[0];
    const int*   ei       = (const int*)  d_in[1];
    const float* W1       = (const float*)d_in[2];
    const float* att_src1 = (const float*)d_in[3];
    const float* att_dst1 = (const float*)d_in[4];
    const float* b1       = (const float*)d_in[5];
    const float* W2       = (const float*)d_in[6];
    const float* att_src2 = (const float*)d_in[7];
    const float* att_dst2 = (const float*)d_in[8];
    const float* b2       = (const float*)d_in[9];
    float* dout = (float*)d_out;

    // workspace carve-up (floats)
    float*    h1     = (float*)d_ws;                       // 6,400,000
    float*    out1   = h1 + (size_t)N_NODES * D1;          // 6,400,000
    float*    asrc1  = out1 + (size_t)N_NODES * D1;        // 200,000
    float*    adst1  = asrc1 + N_NODES * HEADS;            // 200,000
    unsigned* nmax1  = (unsigned*)(adst1 + N_NODES * HEADS);
    float*    denom1 = (float*)(nmax1 + N_NODES * HEADS);
    float*    elog1  = denom1 + N_NODES * HEADS;           // 3,400,000
    float*    h2     = elog1 + (size_t)E_TOT * HEADS;      // 2,000,000
    float*    asrc2  = h2 + (size_t)N_NODES * OUT_DIM;     // 50,000
    float*    adst2  = asrc2 + N_NODES;
    unsigned* nmax2  = (unsigned*)(adst2 + N_NODES);
    float*    denom2 = (float*)(nmax2 + N_NODES);
    float*    elog2  = denom2 + N_NODES;                   // 850,000

    const int T = 256;
    // ---- layer 1 ----
    init1_kernel<<<(N_NODES * D1 + T - 1) / T, T, 0, stream>>>(out1, b1, nmax1, denom1);
    gemm1_wmma<<<N_NODES / 16, 256, 0, stream>>>(x, W1, h1);
    alpha1_kernel<<<(N_NODES * HEADS + T - 1) / T, T, 0, stream>>>(h1, att_src1, att_dst1, asrc1, adst1);
    edge_logits1<<<(E_TOT + T - 1) / T, T, 0, stream>>>(ei, asrc1, adst1, elog1, nmax1);
    edge_exp1<<<(E_TOT + T - 1) / T, T, 0, stream>>>(ei, nmax1, elog1, denom1);
    {
        long long n = (long long)E_TOT * 32;
        edge_msg1<<<(unsigned)((n + T - 1) / T), T, 0, stream>>>(ei, h1, elog1, denom1, out1);
    }
    // ---- layer 2 ----
    init2_kernel<<<(N_NODES * OUT_DIM + T - 1) / T, T, 0, stream>>>(dout, b2, nmax2, denom2);
    gemm2_wmma<<<N_NODES / 16, 96, 0, stream>>>(out1, W2, h2);
    alpha2_kernel<<<(N_NODES + T - 1) / T, T, 0, stream>>>(h2, att_src2, att_dst2, asrc2, adst2);
    edge_logits2<<<(E_TOT + T - 1) / T, T, 0, stream>>>(ei, asrc2, adst2, elog2, nmax2);
    edge_exp2<<<(E_TOT + T - 1) / T, T, 0, stream>>>(ei, nmax2, elog2, denom2);
    {
        long long n = (long long)E_TOT * OUT_DIM;
        edge_msg2<<<(unsigned)((n + T - 1) / T), T, 0, stream>>>(ei, h2, elog2, denom2, dout);
    }
}